// GCNEncoder_58720792871577
// MI455X (gfx1250) — compile-verified
//
#include <hip/hip_runtime.h>
#include <hip/hip_bf16.h>

// ---------------------------------------------------------------------------
// 2-layer GCN for MI455X (gfx1250, wave32).
//   layer: hlin = X @ W  (fp32 WMMA 16x16x4)
//          hagg[dst] += norm(src,dst) * hlin[src]   (incl. self-loops) + bias
//   ReLU between layers.
// Aggregation is the bandwidth bottleneck (~2.5 GB total, L2-resident
// scatter targets); GEMMs use V_WMMA_F32_16X16X4_F32 to keep fp32 accuracy.
// ---------------------------------------------------------------------------

typedef __attribute__((ext_vector_type(2))) float v2f;
typedef __attribute__((ext_vector_type(8))) float v8f;

#define IN_C 512
#define HID 256
#define OUT_C 128

// ---------------------------------------------------------------------------
// Degree / normalization kernels
// ---------------------------------------------------------------------------
__global__ void k_deg_init(float* __restrict__ deg, int n) {
  int i = blockIdx.x * blockDim.x + threadIdx.x;
  if (i < n) deg[i] = 1.0f;  // self-loop contributes 1
}

__global__ void k_deg_count(float* __restrict__ deg,
                            const int* __restrict__ dst, int e) {
  int i = blockIdx.x * blockDim.x + threadIdx.x;
  if (i < e) atomicAdd(&deg[dst[i]], 1.0f);
}

__global__ void k_deg_rsqrt(float* __restrict__ deg, int n) {
  int i = blockIdx.x * blockDim.x + threadIdx.x;
  if (i < n) deg[i] = rsqrtf(deg[i]);
}

// out[i*C + c] = bias[c]   (pre-seeds the scatter-add accumulator with +b)
__global__ void k_fill_bias(float* __restrict__ out,
                            const float* __restrict__ bias,
                            long long total, int C) {
  long long i = (long long)blockIdx.x * blockDim.x + threadIdx.x;
  long long stride = (long long)gridDim.x * blockDim.x;
  for (; i < total; i += stride) out[i] = bias[(int)(i % C)];
}

__global__ void k_relu(float* __restrict__ buf, long long total) {
  long long i = (long long)blockIdx.x * blockDim.x + threadIdx.x;
  long long stride = (long long)gridDim.x * blockDim.x;
  for (; i < total; i += stride) buf[i] = fmaxf(buf[i], 0.0f);
}

// ---------------------------------------------------------------------------
// fp32 WMMA GEMM:  C[M x NCOLS] = A[M x KDIM] @ B[KDIM x NCOLS]
// One wave computes a 16x64 output strip (4 adjacent 16x16 tiles), reusing
// the A fragment across 4 WMMAs per K-step of 4.
//   A fragment (16x4 f32): lane l (0-15): row m0+l, K = k+0..1
//                          lane l (16-31): row m0+(l-15? l&15), K = k+2..3
//   B fragment (4x16 f32): symmetric split: VGPR0 = rows k+{0|2}, VGPR1 = +1
//   C/D (16x16 f32, 8 VGPRs): VGPR r -> M = r + (lane>=16 ? 8 : 0), N = lane&15
// M and K are exact multiples of tile sizes here, so no tail handling and
// EXEC is all-ones inside every executing wave (WMMA requirement).
// ---------------------------------------------------------------------------
template <int KDIM, int NCOLS, int NGROUPS>
__global__ void k_gemm_wmma_f32(const float* __restrict__ A,
                                const float* __restrict__ B,
                                float* __restrict__ C, int mtiles) {
  const int lane = threadIdx.x & 31;
  const int wib  = threadIdx.x >> 5;
  const int wid  = blockIdx.x * (blockDim.x >> 5) + wib;
  const int totalWaves = mtiles * NGROUPS;
  if (wid >= totalWaves) return;  // wave-uniform guard

  const int mt = wid / NGROUPS;
  const int ng = wid % NGROUPS;
  const int m0 = mt * 16;
  const int n0 = ng * 64;

  const int arow  = m0 + (lane & 15);
  const int khalf = (lane >> 4) * 2;  // 0 for lanes 0-15, 2 for lanes 16-31
  const int bcol  = n0 + (lane & 15);

  const float* __restrict__ ap = A + (long long)arow * KDIM;

  v8f c0 = {}, c1 = {}, c2 = {}, c3 = {};

#pragma unroll 4
  for (int k = 0; k < KDIM; k += 4) {
    v2f a;
    a.x = ap[k + khalf];
    a.y = ap[k + khalf + 1];  // pairs -> single global_load_b64 per lane

    const float* __restrict__ br0 = B + (long long)(k + khalf) * NCOLS + bcol;
    const float* __restrict__ br1 = br0 + NCOLS;
    v2f b0, b1, b2, b3;
    b0.x = br0[0];  b0.y = br1[0];
    b1.x = br0[16]; b1.y = br1[16];
    b2.x = br0[32]; b2.y = br1[32];
    b3.x = br0[48]; b3.y = br1[48];

    c0 = __builtin_amdgcn_wmma_f32_16x16x4_f32(false, a, false, b0, (short)0, c0, false, false);
    c1 = __builtin_amdgcn_wmma_f32_16x16x4_f32(false, a, false, b1, (short)0, c1, false, false);
    c2 = __builtin_amdgcn_wmma_f32_16x16x4_f32(false, a, false, b2, (short)0, c2, false, false);
    c3 = __builtin_amdgcn_wmma_f32_16x16x4_f32(false, a, false, b3, (short)0, c3, false, false);
  }

  const int rbase = m0 + (lane >> 4) * 8;
  const int cc    = lane & 15;
#pragma unroll
  for (int r = 0; r < 8; ++r) {
    float* __restrict__ orow = C + (long long)(rbase + r) * NCOLS + n0 + cc;
    orow[0]  = c0[r];
    orow[16] = c1[r];
    orow[32] = c2[r];
    orow[48] = c3[r];
  }
}

// ---------------------------------------------------------------------------
// Edge aggregation: one wave per (edge | self-loop).
//   out[dst] += dinv[src]*dinv[dst] * hlin[src]
// float4 gathers; scalar f32 atomics land in L2 (output fits in 192MB L2).
// ---------------------------------------------------------------------------
template <int C>
__global__ void k_aggregate(const float* __restrict__ hlin,
                            float* __restrict__ out,
                            const int* __restrict__ src,
                            const int* __restrict__ dst,
                            const float* __restrict__ dinv,
                            int E, int N) {
  const int lane = threadIdx.x & 31;
  long long wid = (long long)blockIdx.x * (blockDim.x >> 5) + (threadIdx.x >> 5);
  const long long wstride = (long long)gridDim.x * (blockDim.x >> 5);
  const long long EV = (long long)E + N;

  for (long long e = wid; e < EV; e += wstride) {  // e is wave-uniform
    int s, d;
    if (e < (long long)E) { s = src[e]; d = dst[e]; }
    else                  { s = d = (int)(e - E); }  // self-loop
    const float norm = dinv[s] * dinv[d];
    const float4* __restrict__ hs = (const float4*)(hlin + (long long)s * C);
    float* __restrict__ od = out + (long long)d * C;
#pragma unroll
    for (int j = lane; j < C / 4; j += 32) {
      float4 v = hs[j];
      atomicAdd(&od[4 * j + 0], v.x * norm);
      atomicAdd(&od[4 * j + 1], v.y * norm);
      atomicAdd(&od[4 * j + 2], v.z * norm);
      atomicAdd(&od[4 * j + 3], v.w * norm);
    }
  }
}

// ---------------------------------------------------------------------------
// Launch
// ---------------------------------------------------------------------------
extern "C" void kernel_launch(void* const* d_in, const int* in_sizes, int n_in,
                              void* d_out, int out_size, void* d_ws, size_t ws_size,
                              hipStream_t stream) {
  (void)n_in; (void)out_size; (void)ws_size;

  const float* x  = (const float*)d_in[0];
  const int*   ei = (const int*)d_in[1];   // [2, E] int indices
  const float* W1 = (const float*)d_in[2];
  const float* b1 = (const float*)d_in[3];
  const float* W2 = (const float*)d_in[4];
  const float* b2 = (const float*)d_in[5];
  float* out = (float*)d_out;

  const int N = in_sizes[0] / IN_C;  // 50000
  const int E = in_sizes[1] / 2;     // 800000
  const int* src = ei;
  const int* dst = ei + E;

  // workspace layout: dinv | bufA (N*HID) | bufB (N*HID)
  char* w = (char*)d_ws;
  size_t dinv_bytes = (((size_t)N * 4) + 255) & ~(size_t)255;
  float* dinv = (float*)w;
  float* bufA = (float*)(w + dinv_bytes);       // hlin1, later hlin2
  float* bufB = bufA + (size_t)N * HID;         // hagg1 / relu'd h1

  const int T = 256;  // 8 waves per block

  // 1) symmetric normalization:  dinv = rsqrt(1 + in_degree)
  k_deg_init<<<(N + T - 1) / T, T, 0, stream>>>(dinv, N);
  k_deg_count<<<(E + T - 1) / T, T, 0, stream>>>(dinv, dst, E);
  k_deg_rsqrt<<<(N + T - 1) / T, T, 0, stream>>>(dinv, N);

  const int mtiles = N / 16;  // 3125 (exact)

  // 2) hlin1 = x @ W1   (waves = mtiles * 4 N-groups of 64 cols)
  {
    int waves = mtiles * 4;
    int blocks = (waves + 7) / 8;
    k_gemm_wmma_f32<IN_C, HID, 4><<<blocks, T, 0, stream>>>(x, W1, bufA, mtiles);
  }

  // 3) hagg1 = scatter(norm * gather(hlin1)) + b1 ; ReLU
  k_fill_bias<<<2048, T, 0, stream>>>(bufB, b1, (long long)N * HID, HID);
  k_aggregate<HID><<<4096, T, 0, stream>>>(bufA, bufB, src, dst, dinv, E, N);
  k_relu<<<2048, T, 0, stream>>>(bufB, (long long)N * HID);

  // 4) hlin2 = h1 @ W2  (reuse bufA; waves = mtiles * 2 N-groups)
  {
    int waves = mtiles * 2;
    int blocks = (waves + 7) / 8;
    k_gemm_wmma_f32<HID, OUT_C, 2><<<blocks, T, 0, stream>>>(bufB, W2, bufA, mtiles);
  }

  // 5) out = scatter(norm * gather(hlin2)) + b2
  k_fill_bias<<<2048, T, 0, stream>>>(out, b2, (long long)N * OUT_C, OUT_C);
  k_aggregate<OUT_C><<<4096, T, 0, stream>>>(bufA, out, src, dst, dinv, E, N);
}